// G4Func_71116068487426
// MI455X (gfx1250) — compile-verified
//
#include <hip/hip_runtime.h>
#include <math.h>

// ---------------------------------------------------------------------------
// G4 angular symmetry functions (ANI-style), MI455X / gfx1250.
//
// Strategy:
//  * Stage the full pos[] (120 KB) and species-class[] (10 KB) tables in LDS
//    (320 KB per WGP on CDNA5). Preferred staging path: ONE TDM
//    tensor_load_to_lds per workgroup (D# built per ISA ch.8), waited with
//    s_wait_tensorcnt; any sub-16B remainder staged with
//    global_load_async_to_lds_b32 (+ s_wait_asynccnt). Fallback when the TDM
//    builtin is unavailable: global_load_async_to_lds_b128 loop. Either way,
//    no VGPR round trip; all random gathers become ds_load.
//  * Stream idx_i/idx_j/idx_k/edge_shift coalesced from HBM with
//    global_prefetch_b8 (__builtin_prefetch) ahead of use.
//  * Early-exit on r^2 >= CUTOFF^2 for any leg: only ~1% of triplets
//    contribute, so transcendentals + the 24 atomicAdds per contributing
//    triplet are negligible; atomics resolve in L2 (output 5.76 MB << 192 MB).
// ---------------------------------------------------------------------------

#define MAXN_LDS   10000            // nodes we can stage in LDS (matches ref)
#define CUTOFF     5.0f
#define CUTOFF2    25.0f
#define PI_OVER_CUT 0.62831853071795864769f   // pi / CUTOFF

typedef unsigned int u32x4 __attribute__((ext_vector_type(4)));
typedef int          i32x4 __attribute__((ext_vector_type(4)));
typedef int          i32x8 __attribute__((ext_vector_type(8)));

// TDM builtin exists on both probed toolchains but with different arity:
//   * therock/amdgpu-toolchain (ships the TDM header): 6 args
//     (uint32x4, int32x8, int32x4, int32x4, int32x8, i32 cpol)
//   * ROCm 7.2 clang-22 (no TDM header): 5 args
//     (uint32x4, int32x8, int32x4, int32x4, i32 cpol)
#if defined(__gfx1250__) && __has_builtin(__builtin_amdgcn_tensor_load_to_lds)
#define HAVE_TDM 1
#if __has_include(<hip/amd_detail/amd_gfx1250_TDM.h>)
#define TDM_SIX_ARGS 1
#else
#define TDM_SIX_ARGS 0
#endif
#else
#define HAVE_TDM 0
#endif

__device__ __forceinline__ float fcut(float r) {
    // caller guarantees r < CUTOFF
    return 0.5f * (cosf(PI_OVER_CUT * r) + 1.0f);
}

// CDNA5 async copies global -> LDS, tracked by ASYNCcnt.
__device__ __forceinline__ void async_g2l_b128(unsigned lds_addr,
                                               const void* gsrc) {
#if defined(__gfx1250__)
    asm volatile("global_load_async_to_lds_b128 %0, %1, off"
                 :: "v"(lds_addr), "v"((unsigned long long)(size_t)gsrc)
                 : "memory");
#else
    (void)lds_addr; (void)gsrc;
#endif
}

__device__ __forceinline__ void async_g2l_b32(unsigned lds_addr,
                                              const void* gsrc) {
#if defined(__gfx1250__)
    asm volatile("global_load_async_to_lds_b32 %0, %1, off"
                 :: "v"(lds_addr), "v"((unsigned long long)(size_t)gsrc)
                 : "memory");
#else
    (void)lds_addr; (void)gsrc;
#endif
}

__device__ __forceinline__ void wait_asynccnt0() {
#if defined(__gfx1250__)
    asm volatile("s_wait_asynccnt 0x0" ::: "memory");
#endif
}

__device__ __forceinline__ void wait_tensorcnt0() {
#if defined(__gfx1250__)
#if __has_builtin(__builtin_amdgcn_s_wait_tensorcnt)
    __builtin_amdgcn_s_wait_tensorcnt(0);
#else
    asm volatile("s_wait_tensorcnt 0x0" ::: "memory");
#endif
#endif
}

#if HAVE_TDM
// One-shot TDM copy of `nelem` 4-byte elements (nelem <= 65535) from global
// memory to LDS. D# layout per CDNA5 ISA ch.8 (2D tensor, tile = whole row,
// no padding / iterate / gather / multicast; groups 2/3 unused for <=2D).
__device__ __forceinline__ void tdm_copy_row_4B(const void* gsrc,
                                                unsigned lds_addr,
                                                unsigned nelem) {
    const unsigned long long ga = (unsigned long long)(size_t)gsrc;
    u32x4 g0;
    g0.x = 0x1u;                                   // count=1, user descriptor
    g0.y = lds_addr;                               // lds_addr (bytes)
    g0.z = (unsigned)(ga & 0xffffffffu);           // global_addr[31:0]
    g0.w = (unsigned)((ga >> 32) & 0x01ffffffu)    // global_addr[56:32]
         | (2u << 30);                             // type = 2 ("image")
    i32x8 g1;
    g1[0] = (int)(2u << 16);                       // data_size = 2 (4 bytes)
    g1[1] = (int)((nelem & 0xffffu) << 16);        // tensor_dim0[15:0]
    g1[2] = (int)(((nelem >> 16) & 0xffffu)        // tensor_dim0[31:16]
         | (1u << 16));                            // tensor_dim1[15:0] = 1
    g1[3] = (int)((nelem & 0xffffu) << 16);        // tile_dim0 = nelem
    g1[4] = 1;                                     // tile_dim1 = 1, tile_dim2 = 0
    g1[5] = (int)nelem;                            // tensor_dim0_stride[31:0]
    g1[6] = 0;                                     // stride0[47:32], stride1[15:0]
    g1[7] = 0;                                     // tensor_dim1_stride[47:16]
    const i32x4 gz4 = {0, 0, 0, 0};
#if TDM_SIX_ARGS
    const i32x8 gz8 = {0, 0, 0, 0, 0, 0, 0, 0};
    __builtin_amdgcn_tensor_load_to_lds(g0, g1, gz4, gz4, gz8, 0);
#else
    __builtin_amdgcn_tensor_load_to_lds(g0, g1, gz4, gz4, 0);
#endif
}
#endif

template <bool USE_LDS>
__global__ __launch_bounds__(256)
void g4_kernel(const float* __restrict__ pos,
               const float* __restrict__ cell, int nCell,
               const int*  __restrict__ z,
               const int*  __restrict__ batch,
               const int*  __restrict__ idx_i,
               const int*  __restrict__ idx_j,
               const int*  __restrict__ idx_k,
               const float* __restrict__ eshift,
               const float* __restrict__ etas,
               const float* __restrict__ zetas,
               const float* __restrict__ lambdas,
               float* __restrict__ out,
               int T, int nN)
{
    __shared__ __align__(16) float         s_pos[USE_LDS ? MAXN_LDS * 3 : 4];
    __shared__               unsigned char s_cls[USE_LDS ? MAXN_LDS     : 4];

    if (USE_LDS) {
        const int nfl = nN * 3;                    // total floats to stage
#if HAVE_TDM
        // TDM DMA of the bulk (16B-aligned) part; one issue per workgroup
        // (wave 0 issues; EXEC is ignored by TDM instructions).
        const int nbulk = nfl & ~3;
        if (threadIdx.x < 32) {
            tdm_copy_row_4B(pos, (unsigned)(size_t)s_pos, (unsigned)nbulk);
            wait_tensorcnt0();                     // wave 0's DMA has landed
        }
        // Remainder floats (nfl % 4) via async-to-LDS b32 (0 iters for the
        // reference sizes; correct for arbitrary nN).
        for (int n = nbulk + (int)threadIdx.x; n < nfl; n += blockDim.x)
            async_g2l_b32((unsigned)(size_t)(s_pos + n), pos + n);
        wait_asynccnt0();
#elif defined(__gfx1250__)
        // Async-to-LDS fallback: 16B chunks, tracked by ASYNCcnt.
        const int nv = nfl >> 2;
        for (int v = threadIdx.x; v < nv; v += blockDim.x) {
            const unsigned laddr = (unsigned)(size_t)(s_pos + 4 * (size_t)v);
            async_g2l_b128(laddr, pos + 4 * (size_t)v);
        }
        for (int n = (nv << 2) + threadIdx.x; n < nfl; n += blockDim.x)
            s_pos[n] = pos[n];
        wait_asynccnt0();
#else
        for (int n = threadIdx.x; n < nfl; n += blockDim.x) s_pos[n] = pos[n];
#endif
        // Species class: 1 -> 0, 6 -> 1, 8 -> 2
        for (int n = threadIdx.x; n < nN; n += blockDim.x) {
            int zz = z[n];
            s_cls[n] = (zz == 1) ? 0 : ((zz == 6) ? 1 : 2);
        }
        __syncthreads();    // staged tables visible workgroup-wide
    }

    // Uniform parameter registers (param index p = e*8 + zt*2 + lb).
    const float eta0 = etas[0],  eta1 = etas[8],  eta2 = etas[16];
    const float zt0  = zetas[0], zt1  = zetas[2], zt2  = zetas[4], zt3 = zetas[6];
    const float lam0 = lambdas[0], lam1 = lambdas[1];
    const float pw0 = exp2f(1.0f - zt0), pw1 = exp2f(1.0f - zt1),
                pw2 = exp2f(1.0f - zt2), pw3 = exp2f(1.0f - zt3);

    // Preload the (single) cell if there is only one.
    float c00 = 0, c01 = 0, c02 = 0, c10 = 0, c11 = 0, c12 = 0,
          c20 = 0, c21 = 0, c22 = 0;
    if (nCell == 1) {
        c00 = cell[0]; c01 = cell[1]; c02 = cell[2];
        c10 = cell[3]; c11 = cell[4]; c12 = cell[5];
        c20 = cell[6]; c21 = cell[7]; c22 = cell[8];
    }

    const int tid    = blockIdx.x * blockDim.x + threadIdx.x;
    const int stride = gridDim.x * blockDim.x;

    for (int t = tid; t < T; t += stride) {
        // Prefetch next iteration's streamed data (global_prefetch_b8;
        // speculative, safe past the end of the arrays).
        __builtin_prefetch(idx_i + t + stride, 0, 0);
        __builtin_prefetch(idx_j + t + stride, 0, 0);
        __builtin_prefetch(idx_k + t + stride, 0, 0);
        __builtin_prefetch(eshift + 3 * (size_t)(t + stride), 0, 0);

        const int i = idx_i[t];
        const int j = idx_j[t];
        const int k = idx_k[t];

        const float ex = eshift[3 * (size_t)t + 0];
        const float ey = eshift[3 * (size_t)t + 1];
        const float ez = eshift[3 * (size_t)t + 2];

        float shx, shy, shz;
        if (nCell == 1) {
            shx = ex * c00 + ey * c10 + ez * c20;
            shy = ex * c01 + ey * c11 + ez * c21;
            shz = ex * c02 + ey * c12 + ez * c22;
        } else {
            const float* C = cell + 9 * (size_t)batch[i];
            shx = ex * C[0] + ey * C[3] + ez * C[6];
            shy = ex * C[1] + ey * C[4] + ez * C[7];
            shz = ex * C[2] + ey * C[5] + ez * C[8];
        }

        float pix, piy, piz, pjx, pjy, pjz, pkx, pky, pkz;
        if (USE_LDS) {
            pix = s_pos[3 * i];     piy = s_pos[3 * i + 1]; piz = s_pos[3 * i + 2];
            pjx = s_pos[3 * j];     pjy = s_pos[3 * j + 1]; pjz = s_pos[3 * j + 2];
            pkx = s_pos[3 * k];     pky = s_pos[3 * k + 1]; pkz = s_pos[3 * k + 2];
        } else {
            pix = pos[3 * i];       piy = pos[3 * i + 1];   piz = pos[3 * i + 2];
            pjx = pos[3 * j];       pjy = pos[3 * j + 1];   pjz = pos[3 * j + 2];
            pkx = pos[3 * k];       pky = pos[3 * k + 1];   pkz = pos[3 * k + 2];
        }

        const float vjx = pjx - pix + shx, vjy = pjy - piy + shy, vjz = pjz - piz + shz;
        const float vkx = pkx - pix + shx, vky = pky - piy + shy, vkz = pkz - piz + shz;

        const float rij2 = vjx * vjx + vjy * vjy + vjz * vjz;
        const float rik2 = vkx * vkx + vky * vky + vkz * vkz;
        const float dx = vkx - vjx, dy = vky - vjy, dz = vkz - vjz;
        const float rjk2 = dx * dx + dy * dy + dz * dz;

        // Any leg outside cutoff -> contribution is exactly zero in the
        // reference (cutoff mask) -> skip all transcendental + atomic work.
        if (rij2 >= CUTOFF2 || rik2 >= CUTOFF2 || rjk2 >= CUTOFF2) continue;

        const float rij = sqrtf(rij2), rik = sqrtf(rik2), rjk = sqrtf(rjk2);
        const float fc3 = fcut(rij) * fcut(rik) * fcut(rjk);
        const float cosv = (vjx * vkx + vjy * vky + vjz * vkz)
                           / (rij * rik + 1e-12f);
        const float s = (rij2 + rik2 + rjk2) * (1.0f / (CUTOFF * CUTOFF));

        const float ee0 = expf(-eta0 * s);
        const float ee1 = expf(-eta1 * s);
        const float ee2 = expf(-eta2 * s);

        const float bA = 1.0f + lam0 * cosv;   // lb = 0
        const float bB = 1.0f + lam1 * cosv;   // lb = 1
        const float pA0 = powf(bA, zt0), pA1 = powf(bA, zt1),
                    pA2 = powf(bA, zt2), pA3 = powf(bA, zt3);
        const float pB0 = powf(bB, zt0), pB1 = powf(bB, zt1),
                    pB2 = powf(bB, zt2), pB3 = powf(bB, zt3);

        int cj, ck;
        if (USE_LDS) { cj = s_cls[j]; ck = s_cls[k]; }
        else {
            int zj = z[j], zk = z[k];
            cj = (zj == 1) ? 0 : ((zj == 6) ? 1 : 2);
            ck = (zk == 1) ? 0 : ((zk == 6) ? 1 : 2);
        }
        // same-species channels 0..2; mixed pairs: (0,1)->3 (0,2)->4 (1,2)->5
        const int c = (cj == ck) ? cj : (2 + cj + ck);

        float* ob = out + ((size_t)c * 24) * (size_t)nN + (size_t)i;
        const float sc = 0.5f * fc3;   // reference divides the segment sum by 2

        const float fe[3] = { ee0 * sc, ee1 * sc, ee2 * sc };
#pragma unroll
        for (int e = 0; e < 3; ++e) {
            const size_t row = (size_t)e * 8;
            const float f = fe[e];
            atomicAdd(ob + (row + 0) * (size_t)nN, pw0 * f * pA0);
            atomicAdd(ob + (row + 1) * (size_t)nN, pw0 * f * pB0);
            atomicAdd(ob + (row + 2) * (size_t)nN, pw1 * f * pA1);
            atomicAdd(ob + (row + 3) * (size_t)nN, pw1 * f * pB1);
            atomicAdd(ob + (row + 4) * (size_t)nN, pw2 * f * pA2);
            atomicAdd(ob + (row + 5) * (size_t)nN, pw2 * f * pB2);
            atomicAdd(ob + (row + 6) * (size_t)nN, pw3 * f * pA3);
            atomicAdd(ob + (row + 7) * (size_t)nN, pw3 * f * pB3);
        }
    }
}

extern "C" void kernel_launch(void* const* d_in, const int* in_sizes, int n_in,
                              void* d_out, int out_size, void* d_ws, size_t ws_size,
                              hipStream_t stream)
{
    const float* pos     = (const float*)d_in[0];
    const float* cell    = (const float*)d_in[1];
    const int*   z       = (const int*)  d_in[2];
    const int*   batch   = (const int*)  d_in[3];
    const int*   idx_i   = (const int*)  d_in[4];
    const int*   idx_j   = (const int*)  d_in[5];
    const int*   idx_k   = (const int*)  d_in[6];
    const float* eshift  = (const float*)d_in[7];
    const float* etas    = (const float*)d_in[8];
    const float* zetas   = (const float*)d_in[9];
    const float* lambdas = (const float*)d_in[10];

    const int nN    = in_sizes[0] / 3;
    const int nCell = in_sizes[1] / 9;
    const int T     = in_sizes[4];

    float* out = (float*)d_out;
    hipMemsetAsync(out, 0, (size_t)out_size * sizeof(float), stream);

    if (T <= 0) return;

    const int threads = 256;
    int blocks = (T + threads - 1) / threads;
    if (blocks > 768) blocks = 768;   // balance staging traffic vs. occupancy

    if (nN <= MAXN_LDS) {
        g4_kernel<true><<<blocks, threads, 0, stream>>>(
            pos, cell, nCell, z, batch, idx_i, idx_j, idx_k, eshift,
            etas, zetas, lambdas, out, T, nN);
    } else {
        g4_kernel<false><<<blocks, threads, 0, stream>>>(
            pos, cell, nCell, z, batch, idx_i, idx_j, idx_k, eshift,
            etas, zetas, lambdas, out, T, nN);
    }
}